// GCNNet_70970039599642
// MI455X (gfx1250) — compile-verified
//
#include <hip/hip_runtime.h>
#include <math.h>

typedef __attribute__((ext_vector_type(2))) float v2f;
typedef __attribute__((ext_vector_type(8))) float v8f;

#define GCN_FIN  512
#define GCN_FHID 16
#define GCN_FOUT 40

// ---------------------------------------------------------------- utilities
__global__ void zero_kernel(float* __restrict__ p, size_t n) {
  size_t i = (size_t)blockIdx.x * blockDim.x + threadIdx.x;
  size_t stride = (size_t)gridDim.x * blockDim.x;
  for (; i < n; i += stride) p[i] = 0.0f;
}

__global__ void deg_kernel(const long long* __restrict__ dst,
                           float* __restrict__ deg, int e_cnt) {
  int e = blockIdx.x * blockDim.x + threadIdx.x;
  if (e < e_cnt) atomicAdd(&deg[(int)dst[e]], 1.0f);
}

__global__ void dinv_kernel(float* __restrict__ deg, int n) {
  int i = blockIdx.x * blockDim.x + threadIdx.x;
  if (i < n) deg[i] = rsqrtf(deg[i] + 1.0f);   // +1 self-loop
}

// ------------------------------------------------- GEMM1: h1 = x @ W1 (512->16)
// One wave per 16-row node tile. f32 WMMA 16x16x4, K-loop of 128 steps.
// A 16x4 f32 layout: lane L holds row M=L%16, K = 2*(L/16)+v in VGPR v.
// B 4x16  f32 layout: lane L holds col N=L%16, K = 2*(L/16)+v in VGPR v.
// D 16x16 f32 layout: VGPR v = row v (lanes 0-15) / row v+8 (lanes 16-31).
__global__ __launch_bounds__(256)
void gemm1_wmma(const float* __restrict__ x, const float* __restrict__ W1,
                float* __restrict__ h1, int n) {
  __shared__ float sW[GCN_FIN * GCN_FHID];           // 32 KB
  for (int i = threadIdx.x; i < GCN_FIN * GCN_FHID; i += 256) sW[i] = W1[i];
  __syncthreads();

  const int wave = threadIdx.x >> 5;
  const int lane = threadIdx.x & 31;
  const int row0 = (blockIdx.x * 8 + wave) * 16;     // wave-uniform
  if (row0 >= n) return;                             // whole wave exits: EXEC stays full

  const int m  = lane & 15;
  const int kh = lane >> 4;                          // 0/1: which K pair

  if (row0 + 16 <= n) {
    // ---------------- fast path: full tile, no masks, no store guards ----------
    const float* xrow = x + (size_t)(row0 + m) * GCN_FIN;
    v8f c = {};
    #pragma unroll 4
    for (int k = 0; k < GCN_FIN; k += 4) {
      v2f a, b;
      a.x = xrow[k + 2 * kh];
      a.y = xrow[k + 2 * kh + 1];
      b.x = sW[(k + 2 * kh)     * GCN_FHID + m];
      b.y = sW[(k + 2 * kh + 1) * GCN_FHID + m];
      c = __builtin_amdgcn_wmma_f32_16x16x4_f32(false, a, false, b,
                                                (short)0, c, false, false);
    }
    #pragma unroll
    for (int v = 0; v < 8; ++v)
      h1[(size_t)(row0 + v + 8 * kh) * GCN_FHID + m] = c[v];
  } else {
    // ---------------- tail tile: mask OOB rows to zero -------------------------
    const int arow = row0 + m;
    const float amask = (arow < n) ? 1.0f : 0.0f;
    const float* xrow = x + (size_t)((arow < n) ? arow : 0) * GCN_FIN;
    v8f c = {};
    #pragma unroll 4
    for (int k = 0; k < GCN_FIN; k += 4) {
      v2f a, b;
      a.x = xrow[k + 2 * kh]     * amask;
      a.y = xrow[k + 2 * kh + 1] * amask;
      b.x = sW[(k + 2 * kh)     * GCN_FHID + m];
      b.y = sW[(k + 2 * kh + 1) * GCN_FHID + m];
      c = __builtin_amdgcn_wmma_f32_16x16x4_f32(false, a, false, b,
                                                (short)0, c, false, false);
    }
    #pragma unroll
    for (int v = 0; v < 8; ++v) {
      int r = row0 + v + 8 * kh;
      if (r < n) h1[(size_t)r * GCN_FHID + m] = c[v];
    }
  }
}

// ----------------------------- layer-1 edge aggregation: atomics, L2-resident
__global__ __launch_bounds__(256)
void edge_agg16(const long long* __restrict__ src, const long long* __restrict__ dst,
                const float* __restrict__ dinv, const float* __restrict__ h,
                float* __restrict__ agg, int e_cnt) {
  int e = blockIdx.x * blockDim.x + threadIdx.x;
  if (e >= e_cnt) return;
  int s = (int)src[e], d = (int)dst[e];
  float w = dinv[s] * dinv[d];
  const float4* hr = (const float4*)(h + (size_t)s * GCN_FHID);
  float* ar = agg + (size_t)d * GCN_FHID;
  #pragma unroll
  for (int q = 0; q < 4; ++q) {
    float4 v = hr[q];
    atomicAdd(ar + 4 * q + 0, v.x * w);
    atomicAdd(ar + 4 * q + 1, v.y * w);
    atomicAdd(ar + 4 * q + 2, v.z * w);
    atomicAdd(ar + 4 * q + 3, v.w * w);
  }
}

// ----------------------- self-loop + bias + ReLU (in place into h1) -----------
__global__ __launch_bounds__(256)
void combine1(float* __restrict__ h1, const float* __restrict__ agg1,
              const float* __restrict__ dinv, const float* __restrict__ b1, int n) {
  int gid = blockIdx.x * blockDim.x + threadIdx.x;
  if (gid >= n * GCN_FHID) return;
  int i = gid / GCN_FHID, f = gid % GCN_FHID;
  float di = dinv[i];
  float v = agg1[gid] + h1[gid] * di * di + b1[f];
  h1[gid] = fmaxf(v, 0.0f);
}

// ------------------------------------------- GEMM2: h2 = h1r @ W2 (16->40)
// K=16 -> 4 WMMA steps; N=40 -> 3 column tiles (W2 zero-padded to 16x48 in LDS).
__global__ __launch_bounds__(256)
void gemm2_wmma(const float* __restrict__ h1, const float* __restrict__ W2,
                float* __restrict__ h2, int n) {
  __shared__ float sW[16 * 48];
  for (int i = threadIdx.x; i < 16 * 48; i += 256) {
    int r = i / 48, cc = i % 48;
    sW[i] = (cc < GCN_FOUT) ? W2[r * GCN_FOUT + cc] : 0.0f;
  }
  __syncthreads();

  const int wave = threadIdx.x >> 5;
  const int lane = threadIdx.x & 31;
  const int row0 = (blockIdx.x * 8 + wave) * 16;
  if (row0 >= n) return;

  const int m  = lane & 15;
  const int kh = lane >> 4;
  const bool full = (row0 + 16 <= n);

  // A fragments (K=16): reused across the 3 column tiles
  v2f a[4];
  if (full) {
    const float* hrow = h1 + (size_t)(row0 + m) * GCN_FHID;
    #pragma unroll
    for (int j = 0; j < 4; ++j) {
      a[j].x = hrow[4 * j + 2 * kh];
      a[j].y = hrow[4 * j + 2 * kh + 1];
    }
  } else {
    const int arow = row0 + m;
    const float amask = (arow < n) ? 1.0f : 0.0f;
    const float* hrow = h1 + (size_t)((arow < n) ? arow : 0) * GCN_FHID;
    #pragma unroll
    for (int j = 0; j < 4; ++j) {
      a[j].x = hrow[4 * j + 2 * kh]     * amask;
      a[j].y = hrow[4 * j + 2 * kh + 1] * amask;
    }
  }

  #pragma unroll
  for (int ct = 0; ct < 3; ++ct) {
    const int c0 = ct * 16;
    v8f c = {};
    #pragma unroll
    for (int j = 0; j < 4; ++j) {
      v2f b;
      int k = 4 * j;
      b.x = sW[(k + 2 * kh)     * 48 + c0 + m];
      b.y = sW[(k + 2 * kh + 1) * 48 + c0 + m];
      c = __builtin_amdgcn_wmma_f32_16x16x4_f32(false, a[j], false, b,
                                                (short)0, c, false, false);
    }
    const int col = c0 + m;
    const bool col_ok = (col < GCN_FOUT);
    if (full) {
      #pragma unroll
      for (int v = 0; v < 8; ++v)
        if (col_ok) h2[(size_t)(row0 + v + 8 * kh) * GCN_FOUT + col] = c[v];
    } else {
      #pragma unroll
      for (int v = 0; v < 8; ++v) {
        int r = row0 + v + 8 * kh;
        if (r < n && col_ok) h2[(size_t)r * GCN_FOUT + col] = c[v];
      }
    }
  }
}

// ----------------------------- layer-2 edge aggregation directly into d_out
__global__ __launch_bounds__(256)
void edge_agg40(const long long* __restrict__ src, const long long* __restrict__ dst,
                const float* __restrict__ dinv, const float* __restrict__ h,
                float* __restrict__ out, int e_cnt) {
  int e = blockIdx.x * blockDim.x + threadIdx.x;
  if (e >= e_cnt) return;
  int s = (int)src[e], d = (int)dst[e];
  float w = dinv[s] * dinv[d];
  const float4* hr = (const float4*)(h + (size_t)s * GCN_FOUT);
  float* ar = out + (size_t)d * GCN_FOUT;
  #pragma unroll
  for (int q = 0; q < 10; ++q) {
    float4 v = hr[q];
    atomicAdd(ar + 4 * q + 0, v.x * w);
    atomicAdd(ar + 4 * q + 1, v.y * w);
    atomicAdd(ar + 4 * q + 2, v.z * w);
    atomicAdd(ar + 4 * q + 3, v.w * w);
  }
}

// ------------- self-loop + b2 + log_softmax (in place on d_out), 1 thread/row
__global__ __launch_bounds__(256)
void finalize(const float* __restrict__ h2, const float* __restrict__ dinv,
              const float* __restrict__ b2, float* __restrict__ out, int n) {
  int i = blockIdx.x * blockDim.x + threadIdx.x;
  if (i >= n) return;
  float di = dinv[i];
  float di2 = di * di;
  float v[GCN_FOUT];
  float mx = -INFINITY;
  #pragma unroll
  for (int f = 0; f < GCN_FOUT; ++f) {
    v[f] = out[(size_t)i * GCN_FOUT + f] + h2[(size_t)i * GCN_FOUT + f] * di2 + b2[f];
    mx = fmaxf(mx, v[f]);
  }
  float sum = 0.0f;
  #pragma unroll
  for (int f = 0; f < GCN_FOUT; ++f) sum += expf(v[f] - mx);
  float lse = mx + logf(sum);
  #pragma unroll
  for (int f = 0; f < GCN_FOUT; ++f) out[(size_t)i * GCN_FOUT + f] = v[f] - lse;
}

// ---------------------------------------------------------------- launcher
extern "C" void kernel_launch(void* const* d_in, const int* in_sizes, int n_in,
                              void* d_out, int out_size, void* d_ws, size_t ws_size,
                              hipStream_t stream) {
  const float*     x  = (const float*)d_in[0];
  const long long* ei = (const long long*)d_in[1];   // int64 edge_index [2, E]
  const float*     W1 = (const float*)d_in[2];
  const float*     b1 = (const float*)d_in[3];
  const float*     W2 = (const float*)d_in[4];
  const float*     b2 = (const float*)d_in[5];

  const int n     = in_sizes[0] / GCN_FIN;
  const int e_cnt = in_sizes[1] / 2;
  const long long* src = ei;
  const long long* dst = ei + e_cnt;

  // workspace layout (floats): dinv | agg1 | h1 | h2  (~29.2 MB total)
  float* ws   = (float*)d_ws;
  float* dinv = ws;
  float* agg1 = dinv + n;
  float* h1   = agg1 + (size_t)n * GCN_FHID;
  float* h2   = h1   + (size_t)n * GCN_FHID;
  float* out  = (float*)d_out;

  const int TB = 256;
  const int tiles  = (n + 15) / 16;
  const int gblk   = (tiles + 7) / 8;
  const int eblk   = (e_cnt + TB - 1) / TB;
  const int nblk   = (n + TB - 1) / TB;
  const int nfblk  = ((n * GCN_FHID) + TB - 1) / TB;

  // dinv & agg1 are contiguous -> single zero pass; out zeroed for atomics
  zero_kernel<<<2048, TB, 0, stream>>>(dinv, (size_t)n * (1 + GCN_FHID));
  zero_kernel<<<2048, TB, 0, stream>>>(out, (size_t)n * GCN_FOUT);

  deg_kernel <<<eblk, TB, 0, stream>>>(dst, dinv, e_cnt);
  dinv_kernel<<<nblk, TB, 0, stream>>>(dinv, n);

  gemm1_wmma <<<gblk, TB, 0, stream>>>(x, W1, h1, n);
  edge_agg16 <<<eblk, TB, 0, stream>>>(src, dst, dinv, h1, agg1, e_cnt);
  combine1   <<<nfblk, TB, 0, stream>>>(h1, agg1, dinv, b1, n);

  gemm2_wmma <<<gblk, TB, 0, stream>>>(h1, W2, h2, n);
  edge_agg40 <<<eblk, TB, 0, stream>>>(src, dst, dinv, h2, out, e_cnt);
  finalize   <<<nblk, TB, 0, stream>>>(h2, dinv, b2, out, n);
}